// ChamferLoss_46557445488912
// MI455X (gfx1250) — compile-verified
//
#include <hip/hip_runtime.h>

typedef __attribute__((ext_vector_type(2))) float v2f;
typedef __attribute__((ext_vector_type(8))) float v8f;

#define NPTS 4096
#define NB 8
#define STRIPS (NPTS / 16)   // 256 row-strips per batch per direction

// One wave32 per (dir, batch, 16-row strip). K=4 layout packs the trick:
//   A row m  = [-2*x0, -2*x1, -2*x2, 1]        (pad lane K=3 carries a 1)
//   B col n  = [ y0,    y1,    y2,   |y_n|^2 ] (pad lane K=3 carries y^2)
//   C row v  = |x_row|^2                        (loop-invariant!)
// so a single v_wmma_f32_16x16x4_f32 per tile yields the full squared
// distance D = x^2 + y^2 - 2*x.y with zero per-tile C-build VALU.
// The >=0 clamp commutes with min, so it is hoisted out of the scan loop.
__global__ void __launch_bounds__(256) chamfer_strip_kernel(
    const float* __restrict__ ori, const float* __restrict__ adv,
    float* __restrict__ ws)
{
  const int wave = (blockIdx.x * blockDim.x + threadIdx.x) >> 5;
  const int lane = threadIdx.x & 31;
  const int dir  = wave / (NB * STRIPS);
  const int rem  = wave % (NB * STRIPS);
  const int b    = rem / STRIPS;
  const int strip = rem % STRIPS;

  const float* X = (dir == 0) ? ori : adv;   // rows  (min over the other cloud)
  const float* Y = (dir == 0) ? adv : ori;   // cols
  X += b * 3 * NPTS;
  Y += b * 3 * NPTS;

  const int m  = lane & 15;          // point index within tile
  const bool hi = (lane >> 4) != 0;  // lanes 16-31 hold K=2,3 / rows 8-15
  const int row0 = strip * 16;

  // ---- A matrix (16x4 f32, 2 VGPRs): a.x holds K=2*hi, a.y holds K=2*hi+1,
  //      M = lane&15. K=3 pad lane carries 1.0 (enables the y^2 fold).
  float xa0 = X[0 * NPTS + row0 + m];
  float xa1 = X[1 * NPTS + row0 + m];
  float xa2 = X[2 * NPTS + row0 + m];
  v2f a;
  a.x = -2.0f * (hi ? xa2 : xa0);
  a.y = hi ? 1.0f : (-2.0f * xa1);

  // Loop-invariant C accumulator: |x|^2 broadcast to D-rows M = v + 8*hi.
  float x2all = xa0 * xa0 + xa1 * xa1 + xa2 * xa2;
  v8f cx;
#pragma unroll
  for (int v = 0; v < 8; ++v)
    cx[v] = __shfl(x2all, (v + (hi ? 8 : 0)));

  float rmin[8];
#pragma unroll
  for (int v = 0; v < 8; ++v) rmin[v] = 3.0e38f;

#pragma unroll 2
  for (int col0 = 0; col0 < NPTS; col0 += 16) {
    // ---- B matrix (4x16 f32, 2 VGPRs): bb.x holds K=2*hi, bb.y holds K=2*hi+1,
    //      N = lane&15. Load all 3 coords (no divergence); K=3 lane gets |y|^2.
    float yc0 = Y[0 * NPTS + col0 + m];
    float yc1 = Y[1 * NPTS + col0 + m];
    float yc2 = Y[2 * NPTS + col0 + m];
    float y2 = yc0 * yc0 + yc1 * yc1 + yc2 * yc2;

    v2f bb;
    bb.x = hi ? yc2 : yc0;
    bb.y = hi ? y2 : yc1;

    // v_wmma_f32_16x16x4_f32 : D = A*B + C = -2*x.y + y^2 + x^2
    v8f d = __builtin_amdgcn_wmma_f32_16x16x4_f32(
        /*neg_a=*/false, a, /*neg_b=*/false, bb,
        /*c_mod=*/(short)0, cx, /*reuse_a=*/false, /*reuse_b=*/false);

#pragma unroll
    for (int v = 0; v < 8; ++v)
      rmin[v] = fminf(rmin[v], d[v]);
  }

  // Hoisted clamp (max(.,0) commutes with min), then row-min across the
  // 16 lanes of each half (rows v+8*hi live per-half).
#pragma unroll
  for (int v = 0; v < 8; ++v) {
    rmin[v] = fmaxf(rmin[v], 0.0f);
#pragma unroll
    for (int mask = 8; mask >= 1; mask >>= 1)
      rmin[v] = fminf(rmin[v], __shfl_xor(rmin[v], mask));
  }
  float s = 0.0f;
#pragma unroll
  for (int v = 0; v < 8; ++v) s += rmin[v];   // low half: rows 0-7, high: rows 8-15
  s += __shfl_xor(s, 16);                     // full 16-row strip sum, all lanes
  if (lane == 0)
    ws[dir * (NB * STRIPS) + b * STRIPS + strip] = s;
}

// Single-wave finalize: d1/d2 means per batch, per-batch max, batch mean.
__global__ void chamfer_finalize_kernel(const float* __restrict__ ws,
                                        float* __restrict__ out)
{
  const int lane = threadIdx.x & 31;
  float d[2 * NB];
#pragma unroll
  for (int c = 0; c < 2 * NB; ++c) {
    float s = 0.0f;
    for (int t = lane; t < STRIPS; t += 32) s += ws[c * STRIPS + t];
#pragma unroll
    for (int mask = 16; mask >= 1; mask >>= 1) s += __shfl_xor(s, mask);
    d[c] = s * (1.0f / (float)NPTS);
  }
  if (lane == 0) {
    float acc = 0.0f;
#pragma unroll
    for (int b = 0; b < NB; ++b) acc += fmaxf(d[b], d[NB + b]);
    out[0] = acc * (1.0f / (float)NB);
  }
}

extern "C" void kernel_launch(void* const* d_in, const int* in_sizes, int n_in,
                              void* d_out, int out_size, void* d_ws, size_t ws_size,
                              hipStream_t stream) {
  const float* ori = (const float*)d_in[0];  // [8, 3, 4096] f32
  const float* adv = (const float*)d_in[1];  // [8, 3, 4096] f32
  float* out = (float*)d_out;                // scalar f32
  float* ws  = (float*)d_ws;                 // 2*8*256 floats = 16 KB partials

  // 2 dirs * 8 batches * 256 strips = 4096 waves; 8 waves (256 thr) per block.
  const int total_waves = 2 * NB * STRIPS;
  const int block = 256;
  const int grid = (total_waves * 32) / block;   // 512 blocks
  chamfer_strip_kernel<<<grid, block, 0, stream>>>(ori, adv, ws);
  chamfer_finalize_kernel<<<1, 32, 0, stream>>>(ws, out);
}